// EMDloss_35424890257894
// MI455X (gfx1250) — compile-verified
//
#include <hip/hip_runtime.h>
#include <math.h>

// ---------------------------------------------------------------------------
// EMD approx-match (Fan et al.) for B=8, n=m=2048 3-D point clouds.
//
// d2_ij = |gen_i|^2 + |gt_j|^2 - 2 gen_i . gt_j  computed per 16x16 tile with
// a single V_WMMA_F32_16X16X4_F32:
//   A (16x4, M=gen rows)  : [-2x, -2y, -2z, 1]
//   B (4x16, N=gt cols)   : [ x,   y,   z, |gt|^2]  (pre-baked in lane layout)
//   C (16x16 bias)        : |gen_i|^2 broadcast along rows
// Row/col passes fold lvl2 = level*log2e into the (hoisted) A and C operands,
// so the WMMA emits lvl2*d2 directly and each element needs only v_exp+v_fmac.
// Working set (~1 MB) stays in L2; the kernel is trans/VALU bound.
// ---------------------------------------------------------------------------

#define BATCH   8
#define NPTS    2048
#define TIL     16
#define STRIPS  (NPTS / TIL)          // 128 strips per batch
#define BLOCK   256                   // 8 waves of 32
#define NLEVELS 10
#define LOG2E   1.44269504088896340736f

typedef __attribute__((ext_vector_type(2))) float v2f;
typedef __attribute__((ext_vector_type(8))) float v8f;

// Sum across the 16 lanes of each half-wave (wave32): masks 1,2,4,8.
__device__ __forceinline__ float half_reduce16(float x) {
    x += __shfl_xor(x, 1, 32);
    x += __shfl_xor(x, 2, 32);
    x += __shfl_xor(x, 4, 32);
    x += __shfl_xor(x, 8, 32);
    return x;
}

// One 16x16 d2 tile via f32 WMMA (K=4). B already in lane layout.
__device__ __forceinline__ v8f d2_tile(v2f a, v2f bv, v8f cbias) {
    return __builtin_amdgcn_wmma_f32_16x16x4_f32(
        /*neg_a=*/false, a, /*neg_b=*/false, bv,
        /*c_mod=*/(short)0, cbias, /*reuse_a=*/false, /*reuse_b=*/false);
}

// ---------------------------------------------------------------------------
// Setup: pack A / C-bias, bake B into WMMA lane layout, init remain buffers.
// B lane layout (4x16 KxN, mirroring the documented A 16x4 layout):
//   lanes 0-15 hold {K0,K1} = {x,y};  lanes 16-31 hold {K2,K3} = {z,|gt|^2}
// for column j = tile*16 + (lane&15).
__global__ void k_setup(const float* __restrict__ gen, const float* __restrict__ gt,
                        float4* __restrict__ Ap, float2* __restrict__ Bl,
                        float* __restrict__ nA,
                        float* __restrict__ rl, float* __restrict__ rr,
                        float* __restrict__ costp) {
    int idx = blockIdx.x * blockDim.x + threadIdx.x;
    if (idx < BATCH * NPTS) {
        float x = gen[idx * 3 + 0], y = gen[idx * 3 + 1], z = gen[idx * 3 + 2];
        Ap[idx] = make_float4(-2.0f * x, -2.0f * y, -2.0f * z, 1.0f);
        nA[idx] = x * x + y * y + z * z;
        rl[idx] = 1.0f;   // factorl = max(n,m)/n = 1
        rr[idx] = 1.0f;   // factorr = 1
    }
    if (idx < BATCH * STRIPS * 32) {     // 32768 lane-layout B entries
        int b    = idx / (STRIPS * 32);
        int rem  = idx % (STRIPS * 32);
        int tile = rem >> 5;
        int l    = rem & 31, hl = l >> 4, nl = l & 15;
        int j    = tile * TIL + nl;
        const float* p = gt + (b * NPTS + j) * 3;
        float u = p[0], v = p[1], w = p[2];
        float n2 = u * u + v * v + w * w;
        Bl[idx] = hl ? make_float2(w, n2) : make_float2(u, v);
    }
    if (idx < BATCH * STRIPS) costp[idx] = 0.0f;
}

// ---------------------------------------------------------------------------
// Pass 1: per row i, s_i = sum_j exp(level*d2_ij)*remainr_j ; store
//         f_i = remainl_i / (s_i + 1e-9).   (lvl2 = level * log2e, folded
//         into the hoisted A/C so WMMA yields lvl2*d2 directly.)
__global__ void k_rowpass(float lvl2,
                          const float4* __restrict__ Ap, const float2* __restrict__ Bl,
                          const float* __restrict__ nA,
                          const float* __restrict__ rr, const float* __restrict__ rl,
                          float* __restrict__ f) {
    int wave  = blockIdx.x * (BLOCK / 32) + (threadIdx.x >> 5);
    int b     = wave / STRIPS;
    int strip = wave % STRIPS;
    int l = threadIdx.x & 31, hl = l >> 4, nl = l & 15;
    int i0 = strip * TIL;
    const int base = b * NPTS;
    const float2* BlB = Bl + b * (STRIPS * 32);

    float4 ap = Ap[base + i0 + nl];
    v2f a; a.x = lvl2 * (hl ? ap.z : ap.x); a.y = lvl2 * (hl ? ap.w : ap.y);
    v8f cm;
#pragma unroll
    for (int k = 0; k < 8; ++k) cm[k] = lvl2 * nA[base + i0 + 8 * hl + k];

    float acc[8];
#pragma unroll
    for (int k = 0; k < 8; ++k) acc[k] = 0.0f;

    for (int jt = 0; jt < STRIPS; ++jt) {
        float2 b2 = BlB[jt * 32 + l];
        v2f bv; bv.x = b2.x; bv.y = b2.y;
        float rrv = rr[base + jt * TIL + nl];     // this lane's column
        v8f t = d2_tile(a, bv, cm);               // = lvl2 * d2
#pragma unroll
        for (int k = 0; k < 8; ++k)
            acc[k] += __builtin_amdgcn_exp2f(t[k]) * rrv;
    }
#pragma unroll
    for (int k = 0; k < 8; ++k) {
        float s = half_reduce16(acc[k]);
        if (nl == 0) {
            int i = i0 + 8 * hl + k;
            f[base + i] = rl[base + i] / (s + 1e-9f);
        }
    }
}

// ---------------------------------------------------------------------------
// Pass 2 (+ fused ratio): c_j = remainr_j * sum_i exp(level*d2_ij)*f_i,
//   ratio_j = min(remainr_j/(c_j+1e-9),1),  g_j = remainr_j*ratio_j,
//   remainr_new_j = max(remainr_j - c_j*ratio_j, 0).
// Each column strip is wholly owned by one wave, so the ratio step is local.
__global__ void k_colpass(float lvl2,
                          const float4* __restrict__ Ap, const float2* __restrict__ Bl,
                          const float* __restrict__ nA,
                          const float* __restrict__ rr, const float* __restrict__ f,
                          float* __restrict__ g, float* __restrict__ rr_new) {
    int wave  = blockIdx.x * (BLOCK / 32) + (threadIdx.x >> 5);
    int b     = wave / STRIPS;
    int strip = wave % STRIPS;
    int l = threadIdx.x & 31, hl = l >> 4, nl = l & 15;
    int j0 = strip * TIL;
    const int base = b * NPTS;

    float2 b2 = Bl[b * (STRIPS * 32) + strip * 32 + l];   // hoisted B strip
    v2f bv; bv.x = b2.x; bv.y = b2.y;
    float rrv = rr[base + j0 + nl];

    float coltot = 0.0f;
    for (int i0 = 0; i0 < NPTS; i0 += TIL) {
        float4 ap = Ap[base + i0 + nl];
        v2f a; a.x = lvl2 * (hl ? ap.z : ap.x); a.y = lvl2 * (hl ? ap.w : ap.y);
        v8f cm;
#pragma unroll
        for (int k = 0; k < 8; ++k) cm[k] = lvl2 * nA[base + i0 + 8 * hl + k];
        v8f t = d2_tile(a, bv, cm);               // = lvl2 * d2
#pragma unroll
        for (int k = 0; k < 8; ++k)
            coltot += __builtin_amdgcn_exp2f(t[k]) * f[base + i0 + 8 * hl + k];
    }
    coltot *= rrv;
    coltot += __shfl_xor(coltot, 16, 32);        // fold the two half-waves
    if (l < 16) {
        float cv = coltot;
        float ratio = fminf(rrv / (cv + 1e-9f), 1.0f);
        g[base + j0 + l]      = rrv * ratio;
        rr_new[base + j0 + l] = fmaxf(rrv - cv * ratio, 0.0f);
    }
}

// ---------------------------------------------------------------------------
// Pass 3: w'_ij = exp(level*d2)*g_j*f_i; accumulate cost += w'*sqrt(d2),
//         remainl_i = max(remainl_i - sum_j w'_ij, 0).
// Needs raw d2 for sqrt, so the lvl2 scale stays on the VALU here.
__global__ void k_costpass(float lvl2,
                           const float4* __restrict__ Ap, const float2* __restrict__ Bl,
                           const float* __restrict__ nA,
                           const float* __restrict__ g, const float* __restrict__ f,
                           float* __restrict__ rl, float* __restrict__ costp) {
    int wave  = blockIdx.x * (BLOCK / 32) + (threadIdx.x >> 5);
    int b     = wave / STRIPS;
    int strip = wave % STRIPS;
    int l = threadIdx.x & 31, hl = l >> 4, nl = l & 15;
    int i0 = strip * TIL;
    const int base = b * NPTS;
    const float2* BlB = Bl + b * (STRIPS * 32);

    float4 ap = Ap[base + i0 + nl];
    v2f a; a.x = hl ? ap.z : ap.x; a.y = hl ? ap.w : ap.y;
    v8f cm;
    float fk[8];
#pragma unroll
    for (int k = 0; k < 8; ++k) {
        cm[k] = nA[base + i0 + 8 * hl + k];
        fk[k] = f[base + i0 + 8 * hl + k];
    }

    float rsum[8];
#pragma unroll
    for (int k = 0; k < 8; ++k) rsum[k] = 0.0f;
    float cost = 0.0f;

    for (int jt = 0; jt < STRIPS; ++jt) {
        float2 b2 = BlB[jt * 32 + l];
        v2f bv; bv.x = b2.x; bv.y = b2.y;
        float gv = g[base + jt * TIL + nl];
        v8f d2 = d2_tile(a, bv, cm);
#pragma unroll
        for (int k = 0; k < 8; ++k) {
            float w = __builtin_amdgcn_exp2f(lvl2 * d2[k]) * gv * fk[k];
            rsum[k] += w;
            cost    += w * __builtin_amdgcn_sqrtf(fmaxf(d2[k], 1e-20f));
        }
    }
#pragma unroll
    for (int k = 0; k < 8; ++k) {
        float s = half_reduce16(rsum[k]);
        if (nl == 0) {
            int i = i0 + 8 * hl + k;
            rl[base + i] = fmaxf(rl[base + i] - s, 0.0f);
        }
    }
    cost = half_reduce16(cost);
    cost += __shfl_xor(cost, 16, 32);
    if (l == 0) costp[b * STRIPS + strip] += cost;   // one wave owns this slot
}

// ---------------------------------------------------------------------------
// Final deterministic reduction: out = mean_b( cost_b / n ).
__global__ void k_final(const float* __restrict__ costp, float* __restrict__ out) {
    __shared__ float sb[BATCH];
    int t = threadIdx.x;
    if (t < BATCH) {
        float s = 0.0f;
        for (int k = 0; k < STRIPS; ++k) s += costp[t * STRIPS + k];
        sb[t] = s;
    }
    __syncthreads();
    if (t == 0) {
        float tot = 0.0f;
        for (int b = 0; b < BATCH; ++b) tot += sb[b] / (float)NPTS;
        out[0] = tot / (float)BATCH;
    }
}

// ---------------------------------------------------------------------------
extern "C" void kernel_launch(void* const* d_in, const int* in_sizes, int n_in,
                              void* d_out, int out_size, void* d_ws, size_t ws_size,
                              hipStream_t stream) {
    (void)in_sizes; (void)n_in; (void)out_size; (void)ws_size;
    const float* gen = (const float*)d_in[0];   // [8,2048,3] f32
    const float* gt  = (const float*)d_in[1];   // [8,2048,3] f32
    float* out = (float*)d_out;                 // scalar f32

    // Workspace layout (floats): ~900 KB total.
    float* ws = (float*)d_ws;
    const int PN = BATCH * NPTS;
    float4* Ap    = (float4*)(ws);              // PN float4      (65536 f)
    float2* Bl    = (float2*)(ws + PN * 4);     // PN*2 float2    (65536 f)
    float*  nA    = ws + PN * 8;
    float*  rl    = nA + PN;
    float*  rr0   = rl + PN;
    float*  rr1   = rr0 + PN;
    float*  f     = rr1 + PN;
    float*  g     = f + PN;
    float*  costp = g + PN;                     // BATCH*STRIPS floats

    float lvl2[NLEVELS];
    for (int t = 0; t < 9; ++t) lvl2[t] = -powf(4.0f, (float)(7 - t)) * LOG2E;
    lvl2[9] = 0.0f;

    const int setupBlocks = (BATCH * STRIPS * 32 + BLOCK - 1) / BLOCK;  // 128
    const int passBlocks  = (BATCH * STRIPS) / (BLOCK / 32);            // 128

    k_setup<<<setupBlocks, BLOCK, 0, stream>>>(gen, gt, Ap, Bl, nA, rl, rr0, costp);

    float* rrA = rr0;
    float* rrB = rr1;
    for (int t = 0; t < NLEVELS; ++t) {
        float lv = lvl2[t];
        k_rowpass <<<passBlocks, BLOCK, 0, stream>>>(lv, Ap, Bl, nA, rrA, rl, f);
        k_colpass <<<passBlocks, BLOCK, 0, stream>>>(lv, Ap, Bl, nA, rrA, f, g, rrB);
        k_costpass<<<passBlocks, BLOCK, 0, stream>>>(lv, Ap, Bl, nA, g, f, rl, costp);
        float* tmp = rrA; rrA = rrB; rrB = tmp;
    }
    k_final<<<1, 64, 0, stream>>>(costp, out);
}